// PhysicsConstraints_31430570672937
// MI455X (gfx1250) — compile-verified
//
#include <hip/hip_runtime.h>
#include <hip/hip_bf16.h>

typedef float v2f __attribute__((ext_vector_type(2)));
typedef float v8f __attribute__((ext_vector_type(8)));

#define BATCH 8
#define LEN 2048
#define EPSF 1e-8f
#define C1MIN 3.5f
#define C1MEAN 5.0f
#define TILES_DIM (LEN / 16)               // 128 tiles per row/col
#define JTILES 4                           // j-tiles per strip (16x64 strip)
#define STRIPS_PER_ROW (TILES_DIM / JTILES)          // 32
#define STRIPS_PER_BATCH (TILES_DIM * STRIPS_PER_ROW) // 4096
#define NUM_STRIPS (BATCH * STRIPS_PER_BATCH)         // 32768

// Raw v_sqrt_f32 (TRANS op), no library refinement sequence
__device__ __forceinline__ float fast_sqrtf(float x) {
    return __builtin_amdgcn_sqrtf(x);
}

__global__ void zero_acc_kernel(float* __restrict__ acc) {
    if (threadIdx.x < 4) acc[threadIdx.x] = 0.0f;
}

// norms[b*LEN + l] = x^2 + y^2 + z^2
__global__ void norms_kernel(const float* __restrict__ coords,
                             float* __restrict__ norms) {
    int idx = blockIdx.x * blockDim.x + threadIdx.x;
    if (idx < BATCH * LEN) {
        float x = coords[3 * idx + 0];
        float y = coords[3 * idx + 1];
        float z = coords[3 * idx + 2];
        norms[idx] = x * x + y * y + z * z;
    }
}

// bond energy: sum over (B, L-1) of (sqrt(|p_i - p_{i+1}|^2 + eps) - 5)^2
__global__ void bond_kernel(const float* __restrict__ coords,
                            float* __restrict__ acc) {
    int idx = blockIdx.x * blockDim.x + threadIdx.x;
    float local = 0.0f;
    if (idx < BATCH * (LEN - 1)) {
        int b = idx / (LEN - 1);
        int i = idx - b * (LEN - 1);
        const float* p = coords + (b * LEN + i) * 3;
        float dx = p[0] - p[3];
        float dy = p[1] - p[4];
        float dz = p[2] - p[5];
        float d = fast_sqrtf(dx * dx + dy * dy + dz * dz + EPSF);
        float e = d - C1MEAN;
        local = e * e;
    }
    atomicAdd(&acc[0], local);
}

// Arithmetic encoding of TARGET_TABLE[si][sj] (si,sj in [0,4)):
//   (0,3)/(3,0): sum==3, prod==0 -> 5.9
//   (1,2)/(2,1): sum==3, prod==2 -> 5.7
//   (2,3)/(3,2): prod==6        -> 5.8
//   all other combos            -> 0
__device__ __forceinline__ float target_of(int si, int sj) {
    int p = si * sj;
    int s = si + sj;
    return (s == 3) ? ((p == 0) ? 5.9f : 5.7f)
                    : ((p == 6) ? 5.8f : 0.0f);
}

// Main N^2 kernel. Each wave processes a 16x64 strip: one A fragment
// (rows i0..i0+15, K={x,y,z,0}) reused across 4 WMMAs against 4 B fragments.
// sq[i][j] = n_i + n_j - 2*dot(x_i,x_j); dot tiles come from
// V_WMMA_F32_16X16X4_F32 (16x4 * 4x16 Gram tile).
__global__ void __launch_bounds__(256)
pair_kernel(const float* __restrict__ coords,
            const int* __restrict__ seq,
            const float* __restrict__ norms,
            float* __restrict__ acc) {
    const int lane = threadIdx.x & 31;
    const int waveInBlock = threadIdx.x >> 5;
    const int wavesPerBlock = blockDim.x >> 5;
    const int waveId = blockIdx.x * wavesPerBlock + waveInBlock;
    const int numWaves = gridDim.x * wavesPerBlock;

    // ISA 32-bit 16x4 A layout: lanes 0-15 -> M=lane, K={0,1}; lanes 16-31 ->
    // M=lane-16, K={2,3}. B (4x16) mirrored: N=lane&15, same K split.
    const int half = lane >> 4;   // 0: K={0,1}, 1: K={2,3}
    const int m = lane & 15;
    const int k0 = half * 2;

    float steric_acc = 0.0f;
    float bp_acc = 0.0f;
    float bp_cnt = 0.0f;

    for (int s = waveId; s < NUM_STRIPS; s += numWaves) {
        const int b = s >> 12;                    // / STRIPS_PER_BATCH (4096)
        const int rem = s & (STRIPS_PER_BATCH - 1);
        const int ti = rem >> 5;                  // / STRIPS_PER_ROW (32)
        const int tjg = rem & (STRIPS_PER_ROW - 1);
        const int i0 = ti << 4;
        const int j0base = tjg << 6;              // 64-wide strip

        const float* cb = coords + b * LEN * 3;
        const float* nb = norms + b * LEN;
        const int* sb = seq + b * LEN;

        // A fragment: rows i0..i0+15, this lane's K pair (K=3 -> 0 pad)
        const int ar = i0 + m;
        v2f afrag;
        afrag.x = cb[ar * 3 + k0];
        afrag.y = (k0 + 1 < 3) ? cb[ar * 3 + k0 + 1] : 0.0f;

        // i-side row data, hoisted across all 4 j-tiles.
        // C/D layout: VGPR r -> row i0 + r + half*8 for this lane.
        const int ibase = i0 + half * 8;
        float ni_r[8];
        int si_r[8];
#pragma unroll
        for (int r = 0; r < 8; ++r) {
            ni_r[r] = nb[ibase + r];
            si_r[r] = sb[ibase + r];
        }

#pragma unroll
        for (int jt = 0; jt < JTILES; ++jt) {
            const int j0 = j0base + (jt << 4);

            // B fragment: cols j0..j0+15 of X^T
            const int bc = j0 + m;
            v2f bfrag;
            bfrag.x = cb[bc * 3 + k0];
            bfrag.y = (k0 + 1 < 3) ? cb[bc * 3 + k0 + 1] : 0.0f;

            v8f cacc = {};
            cacc = __builtin_amdgcn_wmma_f32_16x16x4_f32(
                false, afrag, false, bfrag, (short)0, cacc, false, false);

            const int j = j0 + m;          // this lane's column (all VGPRs)
            const float nj = nb[j];
            const int sj = sb[j];

#pragma unroll
            for (int r = 0; r < 8; ++r) {
                const int i = ibase + r;
                float sq = fmaxf(ni_r[r] + nj - 2.0f * cacc[r], 0.0f);

                // steric: |i-j| < 2 entries use eps but still contribute
                const int dij = j - i;
                const int adiff = (dij < 0) ? -dij : dij;
                const float ss = (adiff >= 2) ? sq : EPSF;
                const float d = fast_sqrtf(ss);
                const float tt = fmaxf(C1MIN - d, 0.0f);
                steric_acc = fmaf(tt, tt, steric_acc);

                // bp: branchless select, arithmetic table
                const float tgt = target_of(si_r[r], sj);
                const float da = fast_sqrtf(sq + EPSF);
                const float e = da - tgt;
                const float sel =
                    ((dij >= 3) && (tgt > 0.0f)) ? 1.0f : 0.0f;
                bp_acc = fmaf(sel, e * e, bp_acc);
                bp_cnt += sel;
            }
        }
    }

    atomicAdd(&acc[1], steric_acc);
    atomicAdd(&acc[2], bp_acc);
    atomicAdd(&acc[3], bp_cnt);
}

__global__ void finalize_kernel(const float* __restrict__ acc,
                                float* __restrict__ out) {
    if (threadIdx.x == 0 && blockIdx.x == 0) {
        float bond = acc[0] / (float)(BATCH * (LEN - 1));
        // num_pairs = L^2 - (3L - 2), identical for every batch
        float num_pairs = (float)((long long)LEN * LEN - (3 * LEN - 2));
        float steric = acc[1] / ((float)BATCH * (num_pairs + EPSF));
        float cnt = acc[3];
        float bp = (cnt > 0.0f) ? (acc[2] / fmaxf(cnt, 1.0f)) : 0.0f;
        out[0] = bond;
        out[1] = steric;
        out[2] = bp;
        out[3] = bond + steric + bp;
    }
}

extern "C" void kernel_launch(void* const* d_in, const int* in_sizes, int n_in,
                              void* d_out, int out_size, void* d_ws, size_t ws_size,
                              hipStream_t stream) {
    const float* coords = (const float*)d_in[0];   // (8, 2048, 3) f32
    const int* seq = (const int*)d_in[1];          // (8, 2048) i32
    float* out = (float*)d_out;                    // 4 floats

    float* acc = (float*)d_ws;                     // [bond, steric, bp_sum, bp_cnt]
    float* norms = acc + 4;                        // BATCH*LEN floats

    zero_acc_kernel<<<1, 32, 0, stream>>>(acc);
    norms_kernel<<<(BATCH * LEN + 255) / 256, 256, 0, stream>>>(coords, norms);
    bond_kernel<<<(BATCH * (LEN - 1) + 255) / 256, 256, 0, stream>>>(coords, acc);
    // 32768 strips (16x64 each); 1024 blocks * 8 waves = 8192 waves, 4 strips each
    pair_kernel<<<1024, 256, 0, stream>>>(coords, seq, norms, acc);
    finalize_kernel<<<1, 32, 0, stream>>>(acc, out);
}